// MsgMalPredictor_36584531427420
// MI455X (gfx1250) — compile-verified
//
#include <hip/hip_runtime.h>
#include <hip/hip_bf16.h>

// MI455X / gfx1250, wave32. See analysis: precompute per-node hidden states with
// fp32 WMMA (16x16x4), then a memory-bound per-edge gather/add/relu/project pass.

typedef float v2f __attribute__((ext_vector_type(2)));
typedef float v4f __attribute__((ext_vector_type(4)));
typedef float v8f __attribute__((ext_vector_type(8)));

#define DD 128   // hidden dim (fixed by reference)

// ---------------------------------------------------------------------------
// Kernel A: H = x @ W^T + b for all nodes, using V_WMMA_F32_16X16X4_F32.
// grid = (ceil(N/128), 2); blockIdx.y selects (W_src,b_src)->Hsrc or (W_dst,b_dst)->Hdst.
// Each wave owns a 16-row stripe and produces all 128 output columns
// (8 tiles of 16x16 f32 accumulators = 64 VGPRs).
//
// Fragment layouts per CDNA5 ISA (CDNA5 ISA, 32-bit operands, wave32):
//   A (16x4, MxK):  lanes 0-15 -> M=lane, K={0,1} in vgpr{0,1}; lanes 16-31 -> K={2,3}
//   B (4x16, KxN):  vgpr v -> K=v (lanes 0-15), K=v+2 (lanes 16-31); N = lane&15
//   C/D (16x16):    vgpr v -> M=v (lanes 0-15), M=v+8 (lanes 16-31); N = lane&15
// ---------------------------------------------------------------------------
__global__ __launch_bounds__(256) void node_linear_wmma(
    const float* __restrict__ x,
    const float* __restrict__ Wsrc, const float* __restrict__ bsrc,
    const float* __restrict__ Wdst, const float* __restrict__ bdst,
    float* __restrict__ Hsrc, float* __restrict__ Hdst, int N)
{
  const float* W    = (blockIdx.y == 0) ? Wsrc : Wdst;
  const float* bias = (blockIdx.y == 0) ? bsrc : bdst;
  float*       H    = (blockIdx.y == 0) ? Hsrc : Hdst;

  const int lane  = threadIdx.x & 31;
  const int wave  = threadIdx.x >> 5;
  const int mrow  = lane & 15;   // row / col within tile
  const int khalf = lane >> 4;   // which K pair this lane-half holds

  const int row0 = blockIdx.x * 128 + wave * 16;
  int rowA = row0 + mrow;
  if (rowA >= N) rowA = N - 1;               // clamp loads; stores are guarded
  const float* xrow = x + (size_t)rowA * DD + 2 * khalf;

  v8f acc[8];
#pragma unroll
  for (int t = 0; t < 8; ++t) acc[t] = (v8f)0.0f;

#pragma unroll 4
  for (int kk = 0; kk < 32; ++kk) {          // K = 128 in steps of 4
    // A fragment: x[row][4kk + 2*khalf + {0,1}]  (8B-aligned b64 load)
    v2f a = *(const v2f*)(xrow + 4 * kk);
#pragma unroll
    for (int t = 0; t < 8; ++t) {
      // B fragment: Bmat[k][h] = W[h][k]; h = 16t + mrow, k = 4kk + 2*khalf + {0,1}
      const float* wp = W + (size_t)(16 * t + mrow) * DD + 4 * kk + 2 * khalf;
      v2f bf = *(const v2f*)wp;              // 64KB W stays hot in L1/L2
      acc[t] = __builtin_amdgcn_wmma_f32_16x16x4_f32(
          false, a, false, bf, (short)0, acc[t], false, false);
    }
  }

  // Store D-matrix + bias. vgpr v: row = row0 + v + 8*khalf, col = 16t + mrow.
  const int rsh = khalf * 8;
#pragma unroll
  for (int t = 0; t < 8; ++t) {
    const int   h0 = 16 * t + mrow;
    const float bv = bias[h0];
#pragma unroll
    for (int v = 0; v < 8; ++v) {
      const int r = row0 + v + rsh;
      if (r < N) H[(size_t)r * DD + h0] = acc[t][v] + bv;
    }
  }
}

// ---------------------------------------------------------------------------
// Shared per-edge tail: write edge_embs (non-temporal, keep H tables in L2),
// project to 2 scores with a wave32 xor-shuffle reduction.
// ---------------------------------------------------------------------------
__device__ __forceinline__ void edge_tail(
    v4f h, int lane, long long e,
    const float* __restrict__ Wout, const float* __restrict__ bout,
    float* __restrict__ score, float* __restrict__ emb)
{
  __builtin_nontemporal_store(h, (v4f*)(emb + e * DD) + lane);

  const v4f w0 = *((const v4f*)Wout + lane);
  const v4f w1 = *((const v4f*)(Wout + DD) + lane);
  float p0 = h.x * w0.x + h.y * w0.y + h.z * w0.z + h.w * w0.w;
  float p1 = h.x * w1.x + h.y * w1.y + h.z * w1.z + h.w * w1.w;
#pragma unroll
  for (int off = 16; off > 0; off >>= 1) {
    p0 += __shfl_xor(p0, off, 32);
    p1 += __shfl_xor(p1, off, 32);
  }
  if (lane == 0) {
    score[e * 2 + 0] = p0 + bout[0];
    score[e * 2 + 1] = p1 + bout[1];
  }
}

// ---------------------------------------------------------------------------
// Kernel B: one wave per edge. Fully coalesced 512B gathers from the two
// L2-resident H tables, add + ReLU, store + tiny projection.
// ---------------------------------------------------------------------------
__global__ __launch_bounds__(256) void edge_gather_kernel(
    const float* __restrict__ Hsrc, const float* __restrict__ Hdst,
    const int* __restrict__ src, const int* __restrict__ dst,
    const float* __restrict__ Wout, const float* __restrict__ bout,
    float* __restrict__ score, float* __restrict__ emb, long long E)
{
  const int lane = threadIdx.x & 31;
  const long long e = (long long)blockIdx.x * 8 + (threadIdx.x >> 5);
  if (e >= E) return;
  const long long s = src[e];
  const long long d = dst[e];

  v4f hs = *((const v4f*)(Hsrc + s * DD) + lane);
  v4f hd = *((const v4f*)(Hdst + d * DD) + lane);
  v4f h;
  h.x = fmaxf(hs.x + hd.x, 0.0f);
  h.y = fmaxf(hs.y + hd.y, 0.0f);
  h.z = fmaxf(hs.z + hd.z, 0.0f);
  h.w = fmaxf(hs.w + hd.w, 0.0f);

  edge_tail(h, lane, e, Wout, bout, score, emb);
}

// ---------------------------------------------------------------------------
// Fallback (only if workspace < 2*N*D*4 bytes): fused per-edge compute, one
// wave per edge, 4 hidden values per lane. Slow but correct & capture-safe.
// ---------------------------------------------------------------------------
__global__ __launch_bounds__(256) void edge_fused_fallback(
    const float* __restrict__ x,
    const int* __restrict__ src, const int* __restrict__ dst,
    const float* __restrict__ Wsrc, const float* __restrict__ bsrc,
    const float* __restrict__ Wdst, const float* __restrict__ bdst,
    const float* __restrict__ Wout, const float* __restrict__ bout,
    float* __restrict__ score, float* __restrict__ emb, long long E)
{
  const int lane = threadIdx.x & 31;
  const long long e = (long long)blockIdx.x * 8 + (threadIdx.x >> 5);
  if (e >= E) return;
  const long long s = src[e];
  const long long d = dst[e];
  const float* xs = x + s * DD;
  const float* xd = x + d * DD;

  float hv[4];
#pragma unroll
  for (int j = 0; j < 4; ++j) {
    const int hidx = 4 * lane + j;
    float acc = bsrc[hidx] + bdst[hidx];
    const float* ws = Wsrc + (size_t)hidx * DD;
    const float* wd = Wdst + (size_t)hidx * DD;
    for (int k = 0; k < DD; ++k) acc += xs[k] * ws[k] + xd[k] * wd[k];
    hv[j] = fmaxf(acc, 0.0f);
  }
  v4f h; h.x = hv[0]; h.y = hv[1]; h.z = hv[2]; h.w = hv[3];
  edge_tail(h, lane, e, Wout, bout, score, emb);
}

extern "C" void kernel_launch(void* const* d_in, const int* in_sizes, int n_in,
                              void* d_out, int out_size, void* d_ws, size_t ws_size,
                              hipStream_t stream) {
  const float* x    = (const float*)d_in[0];
  const int*   src  = (const int*)d_in[1];
  const int*   dst  = (const int*)d_in[2];
  const float* Wsrc = (const float*)d_in[3];
  const float* bsrc = (const float*)d_in[4];
  const float* Wdst = (const float*)d_in[5];
  const float* bdst = (const float*)d_in[6];
  const float* Wout = (const float*)d_in[7];
  const float* bout = (const float*)d_in[8];

  const int       N = in_sizes[0] / DD;   // 100000
  const long long E = in_sizes[1];        // 1600000
  const int       C = in_sizes[8];        // 2

  float* score = (float*)d_out;                       // [E, C]
  float* emb   = (float*)d_out + (long long)E * C;    // [E, D]

  const long long edge_blocks = (E + 7) / 8;          // 8 edges (waves) / block
  const size_t need = (size_t)2 * (size_t)N * DD * sizeof(float);

  if (ws_size >= need) {
    float* Hsrc = (float*)d_ws;
    float* Hdst = Hsrc + (size_t)N * DD;
    dim3 gA((N + 127) / 128, 2, 1);
    node_linear_wmma<<<gA, 256, 0, stream>>>(x, Wsrc, bsrc, Wdst, bdst, Hsrc, Hdst, N);
    edge_gather_kernel<<<dim3((unsigned)edge_blocks), 256, 0, stream>>>(
        Hsrc, Hdst, src, dst, Wout, bout, score, emb, E);
  } else {
    edge_fused_fallback<<<dim3((unsigned)edge_blocks), 256, 0, stream>>>(
        x, src, dst, Wsrc, bsrc, Wdst, bdst, Wout, bout, score, emb, E);
  }
}